// FCudaFramework_42185168782031
// MI455X (gfx1250) — compile-verified
//
#include <hip/hip_runtime.h>
#include <math.h>
#include <stdint.h>

// ---------------------------------------------------------------------------
// Problem constants (match reference)
// ---------------------------------------------------------------------------
#define BB   8
#define CC   128
#define HH   48
#define WW   48
#define NN   (HH * WW)          // 2304
#define KTOP 1843               // int(2304 * 0.8)
#define RANK (NN - KTOP)        // 461: ascending 0-based index of threshold

typedef __attribute__((ext_vector_type(2))) float v2f;
typedef __attribute__((ext_vector_type(8))) float v8f;

// D = A(16x4 f32) * B(4x16 f32) + C(16x16 f32), wave32 WMMA.
__device__ __forceinline__ v8f wmma_f32_4(v2f a, v2f b, v8f c) {
    // 8 args: (neg_a, A, neg_b, B, c_mod, C, reuse_a, reuse_b)
    return __builtin_amdgcn_wmma_f32_16x16x4_f32(
        false, a, false, b, (short)0, c, false, false);
}

// ---------------------------------------------------------------------------
// CDNA5 async global->LDS copy (ASYNCcnt-tracked, no VGPR data staging).
// lds_off: byte offset into wave's LDS (generic address low 32 bits).
// ---------------------------------------------------------------------------
__device__ __forceinline__ void async_copy_b128(unsigned lds_off, const float* gptr) {
    asm volatile("global_load_async_to_lds_b128 %0, %1, off"
                 :: "v"(lds_off), "v"(gptr)
                 : "memory");
}
__device__ __forceinline__ void wait_async0() {
    asm volatile("s_wait_asynccnt 0x0" ::: "memory");
}
__device__ __forceinline__ unsigned lds_addr_of(const void* p) {
    return (unsigned)(uintptr_t)p;  // LDS aperture: offset lives in addr[31:0]
}

// ---------------------------------------------------------------------------
// Kernel 1: Y = t @ W^T + bias   (t = x raw-reshaped to [B*N, C])
// One block = 16-row M tile; 8 waves cover 128 output channels.
// A tile staged via async-to-LDS (16x128 f32 = 512 x b128 chunks).
// ---------------------------------------------------------------------------
__global__ __launch_bounds__(256) void qkv_kernel(
    const float* __restrict__ x, const float* __restrict__ Wmat,
    const float* __restrict__ bias, float* __restrict__ out)
{
    const int m0   = blockIdx.x * 16;
    const int tid  = threadIdx.x;
    const int wid  = tid >> 5;
    const int lane = tid & 31;
    const int lr   = lane & 15;
    const int hi   = (lane >= 16) ? 1 : 0;
    const int o0   = wid * 16;

    __shared__ float lt[16][132];   // padded stride: 132 % 64 = 4 -> conflict free
#pragma unroll
    for (int j = 0; j < 2; ++j) {
        int cidx = tid + j * 256;            // 512 chunks of 4 floats
        int r = cidx >> 5, c4 = (cidx & 31) << 2;
        async_copy_b128(lds_addr_of(&lt[r][c4]),
                        x + (size_t)(m0 + r) * CC + c4);
    }
    wait_async0();
    __syncthreads();

    v8f acc = {0.f, 0.f, 0.f, 0.f, 0.f, 0.f, 0.f, 0.f};
    for (int k0 = 0; k0 < CC; k0 += 4) {
        v2f a, b;
        a.x = lt[lr][k0 + 2 * hi];
        a.y = lt[lr][k0 + 2 * hi + 1];
        // B[k][n] = W[n][k]  (y = t @ W^T)
        const float* wrow = Wmat + (size_t)(o0 + lr) * CC + k0 + 2 * hi;
        b.x = wrow[0];
        b.y = wrow[1];
        acc = wmma_f32_4(a, b, acc);
    }

    const float bv = bias[o0 + lr];
#pragma unroll
    for (int g = 0; g < 8; ++g) {
        int M = g + 8 * hi;
        out[(size_t)(m0 + M) * CC + o0 + lr] = acc[g] + bv;
    }
}

// ---------------------------------------------------------------------------
// Kernel 2: S = scale * Qb @ Kb^T  for one batch.  S is [N, N] row-major.
// ---------------------------------------------------------------------------
__global__ __launch_bounds__(256) void scores_kernel(
    const float* __restrict__ Qb, const float* __restrict__ Kb,
    float* __restrict__ S, float scale)
{
    const int m0   = blockIdx.x * 16;
    const int tid  = threadIdx.x;
    const int wid  = tid >> 5;
    const int lane = tid & 31;
    const int lr   = lane & 15;
    const int hi   = (lane >= 16) ? 1 : 0;
    const int n0   = blockIdx.y * 128 + wid * 16;

    __shared__ float lq[16][132];
#pragma unroll
    for (int j = 0; j < 2; ++j) {
        int cidx = tid + j * 256;
        int r = cidx >> 5, c4 = (cidx & 31) << 2;
        async_copy_b128(lds_addr_of(&lq[r][c4]),
                        Qb + (size_t)(m0 + r) * CC + c4);
    }
    wait_async0();
    __syncthreads();

    v8f acc = {0.f, 0.f, 0.f, 0.f, 0.f, 0.f, 0.f, 0.f};
    for (int k0 = 0; k0 < CC; k0 += 4) {
        v2f a, b;
        a.x = lq[lr][k0 + 2 * hi];
        a.y = lq[lr][k0 + 2 * hi + 1];
        // B[k][n] = K[n][k]  (dot of Q row with K row)
        const float* krow = Kb + (size_t)(n0 + lr) * CC + k0 + 2 * hi;
        b.x = krow[0];
        b.y = krow[1];
        acc = wmma_f32_4(a, b, acc);
    }

#pragma unroll
    for (int g = 0; g < 8; ++g) {
        int M = g + 8 * hi;
        S[(size_t)(m0 + M) * NN + n0 + lr] = acc[g] * scale;
    }
}

// ---------------------------------------------------------------------------
// Kernel 3: per-row top-k threshold (radix select) + masked softmax in place.
// One 256-thread block per row. Keeps values >= (KTOP-th largest); others -> 0.
// ---------------------------------------------------------------------------
__device__ __forceinline__ unsigned order_key(float f) {
    unsigned b = __float_as_uint(f);
    return b ^ ((b >> 31) ? 0xFFFFFFFFu : 0x80000000u);  // ascending uint order
}

__global__ __launch_bounds__(256) void topk_softmax_kernel(float* __restrict__ S)
{
    float* s = S + (size_t)blockIdx.x * NN;
    const int tid = threadIdx.x;

    __shared__ float    ls[NN];
    __shared__ unsigned hist[256];
    __shared__ unsigned sh_pref;
    __shared__ int      sh_remain;
    __shared__ float    red[256];

    // Row (9 KB) staged via async copy: 576 x b128 chunks.
    for (int j = tid; j < NN / 4; j += 256)
        async_copy_b128(lds_addr_of(&ls[j * 4]), s + j * 4);
    if (tid == 0) { sh_pref = 0u; sh_remain = RANK; }
    wait_async0();
    __syncthreads();

    // MSB-first radix select: find key of ascending-rank RANK element.
    unsigned prefmask = 0u;
    for (int shift = 24; shift >= 0; shift -= 8) {
        hist[tid] = 0u;
        __syncthreads();
        unsigned pref = sh_pref;
        for (int i = tid; i < NN; i += 256) {
            unsigned k = order_key(ls[i]);
            if ((k & prefmask) == pref)
                atomicAdd(&hist[(k >> shift) & 255u], 1u);
        }
        __syncthreads();
        if (tid == 0) {
            int remain  = sh_remain;
            unsigned cum = 0, v = 0;
            for (v = 0; v < 256; ++v) {
                unsigned h = hist[v];
                if (cum + h > (unsigned)remain) break;
                cum += h;
            }
            sh_pref   = pref | (v << shift);
            sh_remain = remain - (int)cum;
        }
        prefmask |= (0xFFu << shift);
        __syncthreads();
    }
    const unsigned thresh = sh_pref;

    // Row max (global max is always kept, since KTOP >= 1).
    float m = -3.4e38f;
    for (int i = tid; i < NN; i += 256) m = fmaxf(m, ls[i]);
    red[tid] = m;
    __syncthreads();
    for (int off = 128; off > 0; off >>= 1) {
        if (tid < off) red[tid] = fmaxf(red[tid], red[tid + off]);
        __syncthreads();
    }
    m = red[0];
    __syncthreads();

    // Denominator over kept entries (masked entries contribute exp(-1e10)≈0).
    float sum = 0.f;
    for (int i = tid; i < NN; i += 256) {
        if (order_key(ls[i]) >= thresh) sum += __expf(ls[i] - m);
    }
    red[tid] = sum;
    __syncthreads();
    for (int off = 128; off > 0; off >>= 1) {
        if (tid < off) red[tid] += red[tid + off];
        __syncthreads();
    }
    const float inv = 1.0f / red[0];

    for (int i = tid; i < NN; i += 256) {
        float p = (order_key(ls[i]) >= thresh) ? __expf(ls[i] - m) * inv : 0.f;
        s[i] = p;
    }
}

// ---------------------------------------------------------------------------
// Kernel 4: out = P @ V  for one batch.  P: [N,N] (sparse probs), V: [N,C].
// Double-buffered async-to-LDS staging of the 16x128 P chunks overlaps the
// inner-dim (2304) data movement with the WMMA pipeline.
// ---------------------------------------------------------------------------
__global__ __launch_bounds__(256) void pv_kernel(
    const float* __restrict__ P, const float* __restrict__ Vb,
    float* __restrict__ out)
{
    const int m0   = blockIdx.x * 16;
    const int tid  = threadIdx.x;
    const int wid  = tid >> 5;
    const int lane = tid & 31;
    const int lr   = lane & 15;
    const int hi   = (lane >= 16) ? 1 : 0;
    const int c0   = wid * 16;

    __shared__ float lp[2][16][132];

    auto stage = [&](int buf, int chunk) {
#pragma unroll
        for (int j = 0; j < 2; ++j) {
            int cidx = tid + j * 256;        // 512 chunks of 4 floats
            int r = cidx >> 5, c4 = (cidx & 31) << 2;
            async_copy_b128(lds_addr_of(&lp[buf][r][c4]),
                            P + (size_t)(m0 + r) * NN + chunk + c4);
        }
    };

    v8f acc = {0.f, 0.f, 0.f, 0.f, 0.f, 0.f, 0.f, 0.f};

    stage(0, 0);
    wait_async0();
    __syncthreads();

    const int NCHUNK = NN / 128;             // 18
    for (int ci = 0; ci < NCHUNK; ++ci) {
        const int cur = ci & 1;
        if (ci + 1 < NCHUNK) stage(cur ^ 1, (ci + 1) * 128);

        const int chunk = ci * 128;
        for (int k0 = 0; k0 < 128; k0 += 4) {
            v2f a, b;
            a.x = lp[cur][lr][k0 + 2 * hi];
            a.y = lp[cur][lr][k0 + 2 * hi + 1];
            int ng = chunk + k0 + 2 * hi;
            b.x = Vb[(size_t)ng * CC + c0 + lr];
            b.y = Vb[(size_t)(ng + 1) * CC + c0 + lr];
            acc = wmma_f32_4(a, b, acc);
        }
        wait_async0();                        // next buffer resident
        __syncthreads();                      // all waves done with cur buffer
    }

#pragma unroll
    for (int g = 0; g < 8; ++g) {
        int M = g + 8 * hi;
        out[(size_t)(m0 + M) * CC + c0 + lr] = acc[g];
    }
}

// ---------------------------------------------------------------------------
// Kernel 5: importance = sigmoid(conv3x3(xs, gate_w) + gate_b)   [B,1,H,W]
// xs is read in NCHW flat order (raw-reshape identity with [B*N,C] layout).
// ---------------------------------------------------------------------------
__global__ __launch_bounds__(256) void gate_kernel(
    const float* __restrict__ xs, const float* __restrict__ gw,
    const float* __restrict__ gb, float* __restrict__ imp)
{
    int idx = blockIdx.x * 256 + threadIdx.x;
    if (idx >= BB * HH * WW) return;
    int b = idx / (HH * WW);
    int hw = idx % (HH * WW);
    int h = hw / WW, w = hw % WW;

    float acc = gb[0];
    for (int ci = 0; ci < CC; ++ci) {
        const float* xc = xs + ((size_t)(b * CC + ci)) * (HH * WW);
        const float* wp = gw + ci * 9;
#pragma unroll
        for (int kh = 0; kh < 3; ++kh) {
#pragma unroll
            for (int kw = 0; kw < 3; ++kw) {
                int gh = h + kh - 1, gww = w + kw - 1;
                if (gh >= 0 && gh < HH && gww >= 0 && gww < WW)
                    acc += xc[gh * WW + gww] * wp[kh * 3 + kw];
            }
        }
    }
    imp[idx] = 1.f / (1.f + __expf(-acc));
}

// ---------------------------------------------------------------------------
// Kernel 6: x2 = xs + conv3x3(xs*imp, ref_w) + ref_b + 0.5*prev
// Block: 16x16 spatial tile for fixed (b, c_out); LDS halo tile per ci.
// ---------------------------------------------------------------------------
__global__ __launch_bounds__(256) void ref_conv_kernel(
    const float* __restrict__ xs, const float* __restrict__ imp,
    const float* __restrict__ refw, const float* __restrict__ refb,
    const float* __restrict__ prev, float* __restrict__ x2)
{
    const int tile = blockIdx.x;          // 0..8 (3x3 grid of 16x16 tiles)
    const int co   = blockIdx.y;
    const int b    = blockIdx.z;
    const int th0  = (tile / 3) * 16, tw0 = (tile % 3) * 16;
    const int ty   = threadIdx.x / 16, tx = threadIdx.x % 16;

    __shared__ float t[18][18];
    float acc = refb[co];
    const float* impb = imp + (size_t)b * HH * WW;

    for (int ci = 0; ci < CC; ++ci) {
        const float* xc = xs + ((size_t)(b * CC + ci)) * (HH * WW);
        for (int i = threadIdx.x; i < 18 * 18; i += 256) {
            int r = i / 18, c = i % 18;
            int gh = th0 + r - 1, gw = tw0 + c - 1;
            float v = 0.f;
            if (gh >= 0 && gh < HH && gw >= 0 && gw < WW)
                v = xc[gh * WW + gw] * impb[gh * WW + gw];
            t[r][c] = v;
        }
        __syncthreads();
        const float* wp = refw + ((size_t)co * CC + ci) * 9;  // scalar (uniform)
        acc += t[ty + 0][tx + 0] * wp[0] + t[ty + 0][tx + 1] * wp[1] + t[ty + 0][tx + 2] * wp[2]
             + t[ty + 1][tx + 0] * wp[3] + t[ty + 1][tx + 1] * wp[4] + t[ty + 1][tx + 2] * wp[5]
             + t[ty + 2][tx + 0] * wp[6] + t[ty + 2][tx + 1] * wp[7] + t[ty + 2][tx + 2] * wp[8];
        __syncthreads();
    }

    int h = th0 + ty, w = tw0 + tx;
    size_t o = ((size_t)(b * CC + co) * HH + h) * WW + w;
    x2[o] = xs[o] + acc + 0.5f * prev[o];
}

// ---------------------------------------------------------------------------
// Kernel 7: out = conv3x3_dilation2(x2, cov_w) + cov_b   (padding 2)
// ---------------------------------------------------------------------------
__global__ __launch_bounds__(256) void cov_conv_kernel(
    const float* __restrict__ x2, const float* __restrict__ covw,
    const float* __restrict__ covb, float* __restrict__ out)
{
    const int tile = blockIdx.x;
    const int co   = blockIdx.y;
    const int b    = blockIdx.z;
    const int th0  = (tile / 3) * 16, tw0 = (tile % 3) * 16;
    const int ty   = threadIdx.x / 16, tx = threadIdx.x % 16;

    __shared__ float t[20][20];
    float acc = covb[co];

    for (int ci = 0; ci < CC; ++ci) {
        const float* xc = x2 + ((size_t)(b * CC + ci)) * (HH * WW);
        for (int i = threadIdx.x; i < 20 * 20; i += 256) {
            int r = i / 20, c = i % 20;
            int gh = th0 + r - 2, gw = tw0 + c - 2;
            float v = 0.f;
            if (gh >= 0 && gh < HH && gw >= 0 && gw < WW)
                v = xc[gh * WW + gw];
            t[r][c] = v;
        }
        __syncthreads();
        const float* wp = covw + ((size_t)co * CC + ci) * 9;
        acc += t[ty + 0][tx + 0] * wp[0] + t[ty + 0][tx + 2] * wp[1] + t[ty + 0][tx + 4] * wp[2]
             + t[ty + 2][tx + 0] * wp[3] + t[ty + 2][tx + 2] * wp[4] + t[ty + 2][tx + 4] * wp[5]
             + t[ty + 4][tx + 0] * wp[6] + t[ty + 4][tx + 2] * wp[7] + t[ty + 4][tx + 4] * wp[8];
        __syncthreads();
    }

    int h = th0 + ty, w = tw0 + tx;
    out[((size_t)(b * CC + co) * HH + h) * WW + w] = acc;
}

// ---------------------------------------------------------------------------
// Host side
// ---------------------------------------------------------------------------
extern "C" void kernel_launch(void* const* d_in, const int* in_sizes, int n_in,
                              void* d_out, int out_size, void* d_ws, size_t ws_size,
                              hipStream_t stream)
{
    (void)in_sizes; (void)n_in; (void)out_size; (void)ws_size;

    const float* x    = (const float*)d_in[0];
    const float* prev = (const float*)d_in[1];
    const float* Wq   = (const float*)d_in[2];
    const float* bq   = (const float*)d_in[3];
    const float* Wk   = (const float*)d_in[4];
    const float* bk   = (const float*)d_in[5];
    const float* Wv   = (const float*)d_in[6];
    const float* bv   = (const float*)d_in[7];
    const float* refw = (const float*)d_in[8];
    const float* refb = (const float*)d_in[9];
    const float* gw   = (const float*)d_in[10];
    const float* gb   = (const float*)d_in[11];
    const float* covw = (const float*)d_in[12];
    const float* covb = (const float*)d_in[13];

    const size_t NC  = (size_t)NN * CC;     // 294912 per batch
    const size_t BNC = (size_t)BB * NC;     // 2359296

    float* ws  = (float*)d_ws;
    float* Q   = ws;                        // [B*N, C]
    float* Kp  = Q  + BNC;                  // [B*N, C]
    float* V   = Kp + BNC;                  // [B*N, C]
    float* S   = V  + BNC;                  // [N, N]  (reused per batch)
    float* xs  = S  + (size_t)NN * NN;      // [B*N, C] == NCHW flat
    float* imp = xs + BNC;                  // [B, H, W]
    float* x2  = imp + (size_t)BB * HH * WW;// [B, C, H, W]
    float* outp = (float*)d_out;

    // 1) Q/K/V projections (WMMA f32 16x16x4, async-to-LDS A staging)
    qkv_kernel<<<dim3(BB * NN / 16), 256, 0, stream>>>(x, Wq, bq, Q);
    qkv_kernel<<<dim3(BB * NN / 16), 256, 0, stream>>>(x, Wk, bk, Kp);
    qkv_kernel<<<dim3(BB * NN / 16), 256, 0, stream>>>(x, Wv, bv, V);

    // 2) Per-batch: scores -> top-k+softmax -> P@V (S buffer reused; stream order
    //    serializes the dependency, keeping ws at ~65 MB, fully L2-resident).
    const float scale = 1.0f / sqrtf((float)CC);
    for (int b = 0; b < BB; ++b) {
        scores_kernel<<<dim3(NN / 16, NN / 128), 256, 0, stream>>>(
            Q + b * NC, Kp + b * NC, S, scale);
        topk_softmax_kernel<<<dim3(NN), 256, 0, stream>>>(S);
        pv_kernel<<<dim3(NN / 16), 256, 0, stream>>>(S, V + b * NC, xs + b * NC);
    }

    // 3) Gate conv + sigmoid (importance applied on the fly in ref conv)
    gate_kernel<<<dim3((BB * HH * WW + 255) / 256), 256, 0, stream>>>(xs, gw, gb, imp);

    // 4) Refinement conv + residual + temporal propagation (fused)
    ref_conv_kernel<<<dim3(9, CC, BB), 256, 0, stream>>>(xs, imp, refw, refb, prev, x2);

    // 5) Dilated coverage conv -> final output
    cov_conv_kernel<<<dim3(9, CC, BB), 256, 0, stream>>>(x2, covw, covb, outp);
}